// GNNEncoder_23278722744809
// MI455X (gfx1250) — compile-verified
//
#include <hip/hip_runtime.h>

typedef __attribute__((ext_vector_type(2))) float v2f;
typedef __attribute__((ext_vector_type(8))) float v8f;

#define N_NODES 50000
#define N_EDGES 800000

// ---------- float atomic max via monotonic int punning ----------
__device__ __forceinline__ void atomicMaxF(float* addr, float val) {
    if (val >= 0.0f) {
        atomicMax((int*)addr, __float_as_int(val));
    } else {
        atomicMin((unsigned int*)addr, __float_as_uint(val));
    }
}

// ---------- fused QKVS GEMM: X (rows x 64) @ [Wq|Wk|Wv|Ws] (64 x 256) + bias ----------
// Block = 256 threads = 8 waves. One block handles 16 rows; each wave does two 16x16
// output tiles, accumulating K=64 via 16 chained v_wmma_f32_16x16x4_f32.
__global__ __launch_bounds__(256)
void gemm_qkvs(const float* __restrict__ X,
               const float* __restrict__ Wq, const float* __restrict__ bq,
               const float* __restrict__ Wk, const float* __restrict__ bk,
               const float* __restrict__ Wv, const float* __restrict__ bv,
               const float* __restrict__ Ws, const float* __restrict__ bs,
               float* __restrict__ QKVS)
{
    __shared__ float xs[16][68];              // pad 64->68 to spread banks
    const int tid  = threadIdx.x;
    const int row0 = blockIdx.x * 16;

    for (int i = tid; i < 16 * 64; i += 256) {
        int r = i >> 6, c = i & 63;
        xs[r][c] = X[(row0 + r) * 64 + c];
    }
    __syncthreads();

    const int wid  = tid >> 5;
    const int lane = tid & 31;
    const int m    = lane & 15;               // A: M index / B: N index / C: N index
    const int kb   = (lane >> 4) << 1;        // K sub-offset: 0 (lanes 0-15), 2 (lanes 16-31)
    const int hi8  = (lane >> 4) << 3;        // C/D: +8 row offset for upper half-wave

    for (int t = wid; t < 16; t += 8) {       // 16 column tiles of 16
        const int col16 = t << 4;
        const float* W; const float* b;
        switch (t >> 2) {
            case 0:  W = Wq; b = bq; break;
            case 1:  W = Wk; b = bk; break;
            case 2:  W = Wv; b = bv; break;
            default: W = Ws; b = bs; break;
        }
        const int wc = (col16 & 63) + m;      // column inside the selected 64-wide W
        v8f acc = {0.f, 0.f, 0.f, 0.f, 0.f, 0.f, 0.f, 0.f};
        #pragma unroll
        for (int k0 = 0; k0 < 64; k0 += 4) {
            v2f a, bm;
            a[0]  = xs[m][k0 + kb];
            a[1]  = xs[m][k0 + kb + 1];
            bm[0] = W[(k0 + kb)     * 64 + wc];
            bm[1] = W[(k0 + kb + 1) * 64 + wc];
            acc = __builtin_amdgcn_wmma_f32_16x16x4_f32(
                      false, a, false, bm, (short)0, acc, false, false);
        }
        const float bias = b[(col16 & 63) + m];
        #pragma unroll
        for (int r = 0; r < 8; ++r) {
            int row = row0 + hi8 + r;         // M = r + 8*(lane>=16)
            QKVS[row * 256 + col16 + m] = acc[r] + bias;
        }
    }
}

// ---------- per-node prep: out = S (skip term), qdotwe = dot(Q, We), init max/denom ----------
__global__ __launch_bounds__(256)
void node_prep(const float* __restrict__ QKVS, const float* __restrict__ We,
               float* __restrict__ outv, float* __restrict__ nodemax,
               float* __restrict__ denom, float* __restrict__ qdotwe, int n)
{
    int node = blockIdx.x * blockDim.x + threadIdx.x;
    if (node >= n) return;
    const float* q = QKVS + (size_t)node * 256;        // Q block: cols 0..63
    const float* s = q + 192;                           // S block: cols 192..255
    float dot = 0.f;
    #pragma unroll 8
    for (int d = 0; d < 64; ++d) {
        dot += q[d] * We[d];
        outv[(size_t)node * 64 + d] = s[d];
    }
    qdotwe[node]  = dot;
    nodemax[node] = __int_as_float(0xff800000);         // -inf
    denom[node]   = 0.f;
}

// ---------- per-edge score + segment max ----------
__global__ __launch_bounds__(256)
void edge_score(const float* __restrict__ QKVS,
                const int* __restrict__ src, const int* __restrict__ dst,
                const float* __restrict__ ea, const float* __restrict__ qdotwe,
                float* __restrict__ scores, float* __restrict__ nodemax, int ne)
{
    int e = blockIdx.x * blockDim.x + threadIdx.x;
    if (e >= ne) return;
    int sN = src[e], dN = dst[e];
    const float4* q = (const float4*)(QKVS + (size_t)dN * 256);       // Q[dst]
    const float4* k = (const float4*)(QKVS + (size_t)sN * 256 + 64);  // K[src]
    float dot = 0.f;
    #pragma unroll
    for (int i = 0; i < 16; ++i) {
        float4 a = q[i], b = k[i];
        dot += a.x * b.x + a.y * b.y + a.z * b.z + a.w * b.w;
    }
    float sc = (dot + ea[e] * qdotwe[dN]) * 0.125f;     // 1/sqrt(64)
    scores[e] = sc;
    atomicMaxF(&nodemax[dN], sc);
}

// ---------- per-edge exp + segment sum ----------
__global__ __launch_bounds__(256)
void edge_exp(float* __restrict__ scores, const int* __restrict__ dst,
              const float* __restrict__ nodemax, float* __restrict__ denom, int ne)
{
    int e = blockIdx.x * blockDim.x + threadIdx.x;
    if (e >= ne) return;
    int dN = dst[e];
    float a = expf(scores[e] - nodemax[dN]);
    scores[e] = a;                                      // reuse buffer: holds exp(s - m)
    atomicAdd(&denom[dN], a);
}

// ---------- per-edge aggregation: out[dst] += alpha * (V[src] + ea*We) ----------
// One thread per (edge, 4-column group) -> coalesced V reads, 4 fp32 atomics each.
__global__ __launch_bounds__(256)
void edge_agg(const float* __restrict__ QKVS, const float* __restrict__ We,
              const int* __restrict__ src, const int* __restrict__ dst,
              const float* __restrict__ ea,
              const float* __restrict__ scores, const float* __restrict__ denom,
              float* __restrict__ outv, int ne)
{
    long long tid = (long long)blockIdx.x * blockDim.x + threadIdx.x;
    if (tid >= (long long)ne * 16) return;
    int e  = (int)(tid >> 4);
    int c4 = ((int)tid & 15) << 2;
    int sN = src[e], dN = dst[e];
    float coeff = scores[e] / denom[dN];
    float w     = ea[e];
    const float4 v  = *(const float4*)(QKVS + (size_t)sN * 256 + 128 + c4);
    const float4 we = *(const float4*)(We + c4);
    float* o = outv + (size_t)dN * 64 + c4;
    atomicAdd(o + 0, coeff * (v.x + w * we.x));
    atomicAdd(o + 1, coeff * (v.y + w * we.y));
    atomicAdd(o + 2, coeff * (v.z + w * we.z));
    atomicAdd(o + 3, coeff * (v.w + w * we.w));
}

// ---------- in-place ReLU ----------
__global__ __launch_bounds__(256)
void relu_k(float* __restrict__ x, int n)
{
    int i = blockIdx.x * blockDim.x + threadIdx.x;
    if (i < n) x[i] = fmaxf(x[i], 0.f);
}

extern "C" void kernel_launch(void* const* d_in, const int* in_sizes, int n_in,
                              void* d_out, int out_size, void* d_ws, size_t ws_size,
                              hipStream_t stream)
{
    const float* x    = (const float*)d_in[0];
    const int*   ei   = (const int*)d_in[1];
    const float* ea   = (const float*)d_in[2];
    const float* Wq1  = (const float*)d_in[3];
    const float* bq1  = (const float*)d_in[4];
    const float* Wk1  = (const float*)d_in[5];
    const float* bk1  = (const float*)d_in[6];
    const float* Wv1  = (const float*)d_in[7];
    const float* bv1  = (const float*)d_in[8];
    const float* We1  = (const float*)d_in[9];
    const float* Ws1  = (const float*)d_in[10];
    const float* bs1  = (const float*)d_in[11];
    const float* Wq2  = (const float*)d_in[12];
    const float* bq2  = (const float*)d_in[13];
    const float* Wk2  = (const float*)d_in[14];
    const float* bk2  = (const float*)d_in[15];
    const float* Wv2  = (const float*)d_in[16];
    const float* bv2  = (const float*)d_in[17];
    const float* We2  = (const float*)d_in[18];
    const float* Ws2  = (const float*)d_in[19];
    const float* bs2  = (const float*)d_in[20];

    const int* srcI = ei;
    const int* dstI = ei + N_EDGES;
    float* out = (float*)d_out;

    // workspace layout (floats)
    float* ws      = (float*)d_ws;
    float* qkvs    = ws;                                   // N*256
    float* out1    = qkvs    + (size_t)N_NODES * 256;      // N*64 (layer-1 result / layer-2 input)
    float* scores  = out1    + (size_t)N_NODES * 64;       // E
    float* nodemax = scores  + N_EDGES;                    // N
    float* denom   = nodemax + N_NODES;                    // N
    float* qdotwe  = denom   + N_NODES;                    // N

    dim3 blk(256);
    const int gGemm = N_NODES / 16;                        // 3125 (N divisible by 16)
    const int gNode = (N_NODES + 255) / 256;
    const int gEdge = (N_EDGES + 255) / 256;
    const int gAgg  = (int)(((long long)N_EDGES * 16 + 255) / 256);
    const int gElem = (N_NODES * 64 + 255) / 256;

    // ---- layer 1 ----
    gemm_qkvs<<<gGemm, blk, 0, stream>>>(x, Wq1, bq1, Wk1, bk1, Wv1, bv1, Ws1, bs1, qkvs);
    node_prep<<<gNode, blk, 0, stream>>>(qkvs, We1, out1, nodemax, denom, qdotwe, N_NODES);
    edge_score<<<gEdge, blk, 0, stream>>>(qkvs, srcI, dstI, ea, qdotwe, scores, nodemax, N_EDGES);
    edge_exp<<<gEdge, blk, 0, stream>>>(scores, dstI, nodemax, denom, N_EDGES);
    edge_agg<<<gAgg, blk, 0, stream>>>(qkvs, We1, srcI, dstI, ea, scores, denom, out1, N_EDGES);
    relu_k<<<gElem, blk, 0, stream>>>(out1, N_NODES * 64);

    // ---- layer 2 ----
    gemm_qkvs<<<gGemm, blk, 0, stream>>>(out1, Wq2, bq2, Wk2, bk2, Wv2, bv2, Ws2, bs2, qkvs);
    node_prep<<<gNode, blk, 0, stream>>>(qkvs, We2, out, nodemax, denom, qdotwe, N_NODES);
    edge_score<<<gEdge, blk, 0, stream>>>(qkvs, srcI, dstI, ea, qdotwe, scores, nodemax, N_EDGES);
    edge_exp<<<gEdge, blk, 0, stream>>>(scores, dstI, nodemax, denom, N_EDGES);
    edge_agg<<<gAgg, blk, 0, stream>>>(qkvs, We2, srcI, dstI, ea, scores, denom, out, N_EDGES);
}